// YOLOXLoss_6073083756634
// MI455X (gfx1250) — compile-verified
//
#include <hip/hip_runtime.h>
#include <hip/hip_bf16.h>
#include <math.h>

// ---------------------------------------------------------------------------
// YOLOX total loss on MI455X (gfx1250, wave32).
// Streaming reduction: ~182MB in -> 1 float out. HBM-bound (23.3 TB/s).
// loss = (5*sum_iou + sum_obj_bce + sum_cls_bce) / max(sum_fg, 1)
// ---------------------------------------------------------------------------

typedef __attribute__((ext_vector_type(2))) float v2f;
typedef __attribute__((ext_vector_type(8))) float v8f;

#define BATCH   32
#define A_TOT   8400          // 6400 + 1600 + 400
#define G_TOT   (BATCH * A_TOT)   // 268800
#define BLK     256
#define NB      (G_TOT / BLK)     // 1050, exact

__device__ __forceinline__ float bce_logits(float x, float t) {
    // max(x,0) - x*t + log1p(exp(-|x|))
    return fmaxf(x, 0.0f) - x * t + log1pf(expf(-fabsf(x)));
}

__global__ __launch_bounds__(BLK)
void yolox_partial(const float* __restrict__ out0,
                   const float* __restrict__ out1,
                   const float* __restrict__ out2,
                   const float* __restrict__ reg_t,
                   const float* __restrict__ cls_t,
                   const int*   __restrict__ fg_mask,
                   float*       __restrict__ partials) {
    const int g = blockIdx.x * BLK + threadIdx.x;   // exact cover of G_TOT
    const int b = g / A_TOT;
    const int a = g - b * A_TOT;

    // Select FPN level. Channel-major layout: obase[c*HW + hw] is coalesced
    // across lanes (consecutive hw).
    const float* obase;
    int HW, W, hw;
    float stride_;
    if (a < 6400)      { obase = out0 + (size_t)b * 85 * 6400; HW = 6400; W = 80; stride_ = 8.0f;  hw = a; }
    else if (a < 8000) { obase = out1 + (size_t)b * 85 * 1600; HW = 1600; W = 40; stride_ = 16.0f; hw = a - 6400; }
    else               { obase = out2 + (size_t)b * 85 * 400;  HW = 400;  W = 20; stride_ = 32.0f; hw = a - 8000; }

    const float fg = (float)fg_mask[g];

    // Objectness BCE is unmasked: always needed.
    float n_local = bce_logits(obase[4 * HW + hw], fg);

    if (fg != 0.0f) {
        // fg==1 lanes only: bbox decode + IoU + 80-class BCE.
        const float* ct = cls_t + (size_t)g * 80;
        __builtin_prefetch(ct, 0, 0);                       // global_prefetch_b8

        // ---- decode ----
        const float px = obase[0 * HW + hw];
        const float py = obase[1 * HW + hw];
        const float pw = obase[2 * HW + hw];
        const float ph = obase[3 * HW + hw];
        const int   gx = hw % W;
        const int   gy = hw / W;
        const float cx = (px + (float)gx) * stride_;
        const float cy = (py + (float)gy) * stride_;
        const float w  = expf(pw) * stride_;
        const float h  = expf(ph) * stride_;

        // ---- IoU loss ----
        const float4 t = *reinterpret_cast<const float4*>(reg_t + (size_t)g * 4);
        const float tlx = fmaxf(cx - 0.5f * w, t.x - 0.5f * t.z);
        const float tly = fmaxf(cy - 0.5f * h, t.y - 0.5f * t.w);
        const float brx = fminf(cx + 0.5f * w, t.x + 0.5f * t.z);
        const float bry = fminf(cy + 0.5f * h, t.y + 0.5f * t.w);
        const float area_p = w * h;
        const float area_g = t.z * t.w;
        const float en     = ((tlx < brx) && (tly < bry)) ? 1.0f : 0.0f;
        const float area_i = (brx - tlx) * (bry - tly) * en;
        const float area_u = area_p + area_g - area_i;
        const float iou    = area_i / (area_u + 1e-16f);
        n_local += 5.0f * (1.0f - iou * iou);               // REG_WEIGHT folded in

        // ---- class BCE (80 classes) ----
        const float4* ct4 = reinterpret_cast<const float4*>(ct);  // 320B aligned
        const float*  oc  = obase + 5 * HW + hw;
        float cls_sum = 0.0f;
        #pragma unroll 5
        for (int k4 = 0; k4 < 20; ++k4) {
            const float4 tt = ct4[k4];
            cls_sum += bce_logits(oc[0 * HW], tt.x);
            cls_sum += bce_logits(oc[1 * HW], tt.y);
            cls_sum += bce_logits(oc[2 * HW], tt.z);
            cls_sum += bce_logits(oc[3 * HW], tt.w);
            oc += 4 * HW;
        }
        n_local += cls_sum;
    }
    float f_local = fg;

    // ---- deterministic block reduction (wave32 shfl tree + LDS) ----
    #pragma unroll
    for (int m = 16; m >= 1; m >>= 1) {
        n_local += __shfl_xor(n_local, m, 32);
        f_local += __shfl_xor(f_local, m, 32);
    }
    __shared__ float sn[BLK / 32], sf[BLK / 32];
    const int lane = threadIdx.x & 31;
    const int wid  = threadIdx.x >> 5;
    if (lane == 0) { sn[wid] = n_local; sf[wid] = f_local; }
    __syncthreads();
    if (threadIdx.x == 0) {
        float n = 0.0f, f = 0.0f;
        #pragma unroll
        for (int i = 0; i < BLK / 32; ++i) { n += sn[i]; f += sf[i]; }
        partials[2 * blockIdx.x + 0] = n;
        partials[2 * blockIdx.x + 1] = f;
    }
}

// Final reduction: one wave. Uses V_WMMA_F32_16X16X4_F32 with A = ones.
// With A == 1, every D[m][n] equals column-sum(B[:,n]); summing ALL 256 D
// entries therefore yields 16 * (sum of all 64 B slots) independent of the
// B-operand lane layout — a layout-agnostic wave reduction.
__global__ __launch_bounds__(32)
void yolox_final(const float* __restrict__ partials, float* __restrict__ out) {
    const int t = threadIdx.x;
    float n = 0.0f, f = 0.0f;
    for (int i = t; i < NB; i += 32) {   // fixed order -> deterministic
        n += partials[2 * i + 0];
        f += partials[2 * i + 1];
    }
    // EXEC is all-ones here (full wave, reconverged).
    v2f ones; ones[0] = 1.0f; ones[1] = 1.0f;
    v2f bn;   bn[0]   = n;    bn[1]   = 0.0f;
    v2f bf;   bf[0]   = f;    bf[1]   = 0.0f;
    v8f cz = {};
    v8f dn = __builtin_amdgcn_wmma_f32_16x16x4_f32(false, ones, false, bn,
                                                   (short)0, cz, false, false);
    v8f df = __builtin_amdgcn_wmma_f32_16x16x4_f32(false, ones, false, bf,
                                                   (short)0, cz, false, false);
    float sn = dn[0] + dn[1] + dn[2] + dn[3] + dn[4] + dn[5] + dn[6] + dn[7];
    float sf = df[0] + df[1] + df[2] + df[3] + df[4] + df[5] + df[6] + df[7];
    #pragma unroll
    for (int m = 16; m >= 1; m >>= 1) {
        sn += __shfl_xor(sn, m, 32);
        sf += __shfl_xor(sf, m, 32);
    }
    if (t == 0) {
        const float num    = sn * (1.0f / 16.0f);
        const float fg_sum = sf * (1.0f / 16.0f);
        out[0] = num / fmaxf(fg_sum, 1.0f);
    }
}

extern "C" void kernel_launch(void* const* d_in, const int* in_sizes, int n_in,
                              void* d_out, int out_size, void* d_ws, size_t ws_size,
                              hipStream_t stream) {
    const float* out0 = (const float*)d_in[0];
    const float* out1 = (const float*)d_in[1];
    const float* out2 = (const float*)d_in[2];
    const float* regt = (const float*)d_in[3];
    const float* clst = (const float*)d_in[4];
    const int*   fgm  = (const int*)  d_in[5];
    float* partials = (float*)d_ws;          // NB*2 floats = 8.4 KB
    float* outp     = (float*)d_out;

    yolox_partial<<<NB, BLK, 0, stream>>>(out0, out1, out2, regt, clst, fgm, partials);
    yolox_final<<<1, 32, 0, stream>>>(partials, outp);
}